// SOAPDescriptor_16819091931699
// MI455X (gfx1250) — compile-verified
//
#include <hip/hip_runtime.h>

// ---------------------------------------------------------------------------
// SOAP descriptor for MI455X (gfx1250, wave32).
//
// Phase 1: edge scatter -> coeffs[atom][type][radial=8][sh=9] f32 (57.6 MB,
//          fits in 192MB L2) using native global_atomic_add_f32.
// Phase 2: per (atom,type) segment, 3 per-slice 8x8 Grams via
//          V_WMMA_F32_16X16X4_F32. Two segments packed per 16x4 A-tile;
//          src0==src1 computes A*A^T because the f32 A and B VGPR layouts
//          are structurally identical (lane=M vs lane=N, same K striping).
// ---------------------------------------------------------------------------

#define SOAP_NUM_TYPES  4
#define SOAP_NUM_RADIAL 8
#define SOAP_SH_DIM     9
#define SOAP_CUTOFF     5.0f
#define SOAP_CPS        (SOAP_NUM_RADIAL * SOAP_SH_DIM)   // 72 floats per segment
#define SOAP_OUT_SEG    (3 * 36)                           // 108 floats per segment

typedef __attribute__((ext_vector_type(2))) float v2f;
typedef __attribute__((ext_vector_type(8))) float v8f;

// ---------------------------------------------------------------------------
// Zero the coeffs accumulator (vectorized b128 stores + tail).
// ---------------------------------------------------------------------------
__global__ void soap_zero_kernel(float* __restrict__ p, int n) {
    int n4 = n >> 2;
    for (int i = blockIdx.x * blockDim.x + threadIdx.x; i < n4;
         i += gridDim.x * blockDim.x) {
        ((float4*)p)[i] = make_float4(0.f, 0.f, 0.f, 0.f);
    }
    if (blockIdx.x == 0 && threadIdx.x < (n & 3)) {
        p[(n4 << 2) + threadIdx.x] = 0.f;
    }
}

// ---------------------------------------------------------------------------
// Phase 1: one lane per edge; 72 native f32 atomic adds into L2-resident
// accumulator.
// ---------------------------------------------------------------------------
__global__ void soap_scatter_kernel(const int*   __restrict__ edge_index,
                                    const float* __restrict__ edge_length,
                                    const float* __restrict__ edge_vec,
                                    const int*   __restrict__ types,
                                    float*       __restrict__ coeffs,
                                    int n_edges) {
    int e = blockIdx.x * blockDim.x + threadIdx.x;
    if (e >= n_edges) return;

    const int   row = edge_index[e];
    const int   col = edge_index[n_edges + e];
    const float len = edge_length[e];
    const float vx = edge_vec[3 * e + 0];
    const float vy = edge_vec[3 * e + 1];
    const float vz = edge_vec[3 * e + 2];

    // direction = vec / max(len, 1e-8); u = direction / max(|direction|, 1e-12)
    const float inv = 1.0f / fmaxf(len, 1e-8f);
    const float dx = vx * inv, dy = vy * inv, dz = vz * inv;
    const float nrm = fmaxf(sqrtf(dx * dx + dy * dy + dz * dz), 1e-12f);
    const float rn = 1.0f / nrm;
    const float x = dx * rn, y = dy * rn, z = dz * rn;

    const float s3  = 1.7320508075688772f;   // sqrt(3)
    const float s5  = 2.2360679774997896f;   // sqrt(5)
    const float s15 = 3.8729833462074170f;   // sqrt(15)

    float sh[SOAP_SH_DIM];
    sh[0] = 1.0f;
    sh[1] = s3 * x;
    sh[2] = s3 * y;
    sh[3] = s3 * z;
    sh[4] = s15 * x * z;
    sh[5] = s15 * x * y;
    sh[6] = s5 * (y * y - 0.5f * (x * x + z * z));
    sh[7] = s15 * y * z;
    sh[8] = 0.5f * s15 * (z * z - x * x);

    // envelope
    const float scaled = len * (1.0f / SOAP_CUTOFF);
    float env = 0.0f;
    if (scaled < 1.0f) {
        const float denom = fmaxf(1.0f - scaled * scaled, 1e-12f);
        env = expf(1.0f - 1.0f / denom);
    }

    // gamma = 1/(CUTOFF/(NUM_RADIAL-1))^2 = (7/5)^2
    const float gamma   = 1.96f;
    const float spacing = SOAP_CUTOFF / 7.0f;

    const int seg = row * SOAP_NUM_TYPES + types[col];
    float* base = coeffs + (size_t)seg * SOAP_CPS;

#pragma unroll
    for (int i = 0; i < SOAP_NUM_RADIAL; ++i) {
        const float d = len - (float)i * spacing;
        const float w = env * expf(-gamma * d * d);
#pragma unroll
        for (int m = 0; m < SOAP_SH_DIM; ++m) {
            unsafeAtomicAdd(base + i * SOAP_SH_DIM + m, w * sh[m]);
        }
    }
}

// ---------------------------------------------------------------------------
// Phase 2: one wave per pair of segments. A-tile 16x4 f32 (rows 0-7 = seg0
// radial rows, rows 8-15 = seg1). src0 == src1 -> D = A*A^T. 4 WMMAs cover
// the 3 sh-slices (m={0}, {1..3}, {4..7}+{8}), zero-padded K columns.
// D layout: lane l holds (M = vgpr + 8*(l>>4), N = l&15). Diagonal 8x8
// blocks are the two Grams; lanes 0-7 extract seg0, lanes 24-31 seg1.
// ---------------------------------------------------------------------------
__global__ void soap_gram_kernel(const float* __restrict__ coeffs,
                                 float* __restrict__ out,
                                 int n_pairs) {
    const int lane = threadIdx.x & 31;
    const int half = lane >> 4;          // K-half for A, M-half for D
    const int r16  = lane & 15;          // A row (M) / D column (N)
    const int wave  = (blockIdx.x * blockDim.x + threadIdx.x) >> 5;
    const int nwave = (gridDim.x * blockDim.x) >> 5;

    for (int pair = wave; pair < n_pairs; pair += nwave) {   // wave-uniform
        const int seg = pair * 2 + (r16 >> 3);
        const float* a = coeffs + (size_t)seg * SOAP_CPS + (r16 & 7) * SOAP_SH_DIM;

        const float m0 = a[0], m1 = a[1], m2 = a[2], m3 = a[3], m4 = a[4];
        const float m5 = a[5], m6 = a[6], m7 = a[7], m8 = a[8];

        // K columns per lane-half; v_cndmask selects, EXEC stays all-1s.
        v2f a0 = { half ? 0.f : m0, 0.f };            // slice0: m=0
        v2f a1 = { half ? m3  : m1, half ? 0.f : m2 };// slice1: m=1,2,3
        v2f a2 = { half ? m6  : m4, half ? m7  : m5 };// slice2a: m=4..7
        v2f a3 = { half ? 0.f : m8, 0.f };            // slice2b: m=8

        v8f c0 = {}, c1 = {}, c2 = {};
        c0 = __builtin_amdgcn_wmma_f32_16x16x4_f32(false, a0, false, a0,
                                                   (short)0, c0, false, false);
        c1 = __builtin_amdgcn_wmma_f32_16x16x4_f32(false, a1, false, a1,
                                                   (short)0, c1, false, false);
        c2 = __builtin_amdgcn_wmma_f32_16x16x4_f32(false, a2, false, a2,
                                                   (short)0, c2, false, false);
        c2 = __builtin_amdgcn_wmma_f32_16x16x4_f32(false, a3, false, a3,
                                                   (short)0, c2, false, false);

        // Upper-triangle extraction (divergent; all WMMAs already issued).
        const bool w0 = (half == 0) && (r16 < 8);    // seg0 block, col = r16
        const bool w1 = (half == 1) && (r16 >= 8);   // seg1 block, col = r16-8
        if (w0 || w1) {
            const int c    = w0 ? r16 : (r16 - 8);
            const int oseg = pair * 2 + (w1 ? 1 : 0);
            float* ob = out + (size_t)oseg * SOAP_OUT_SEG;
#pragma unroll
            for (int r = 0; r < 8; ++r) {
                if (r <= c) {
                    const int t = 8 * r - (r * (r - 1)) / 2 + (c - r); // triu idx
                    ob[0 * 36 + t] = c0[r];
                    ob[1 * 36 + t] = c1[r];
                    ob[2 * 36 + t] = c2[r];
                }
            }
        }
    }
}

// ---------------------------------------------------------------------------
// Inputs (setup_inputs order): edge_index (2*E i32), edge_length (E f32),
// edge_vec (E*3 f32), num_atoms (scalar), types (N i32). Output: f32.
// ---------------------------------------------------------------------------
extern "C" void kernel_launch(void* const* d_in, const int* in_sizes, int n_in,
                              void* d_out, int out_size, void* d_ws, size_t ws_size,
                              hipStream_t stream) {
    const int*   edge_index  = (const int*)d_in[0];
    const float* edge_length = (const float*)d_in[1];
    const float* edge_vec    = (const float*)d_in[2];
    const int*   types       = (const int*)d_in[4];

    const int n_edges = in_sizes[1];
    const int n_atoms = in_sizes[4];
    const int n_coeff = n_atoms * SOAP_NUM_TYPES * SOAP_CPS;  // 14.4M f32
    const int n_pairs = (n_atoms * SOAP_NUM_TYPES) / 2;       // 100k

    float* coeffs = (float*)d_ws;   // 57.6 MB accumulator (L2-resident)

    {
        int blocks = (n_coeff / 4 + 255) / 256;
        soap_zero_kernel<<<blocks, 256, 0, stream>>>(coeffs, n_coeff);
    }
    {
        int blocks = (n_edges + 255) / 256;
        soap_scatter_kernel<<<blocks, 256, 0, stream>>>(
            edge_index, edge_length, edge_vec, types, coeffs, n_edges);
    }
    {
        int blocks = (n_pairs + 7) / 8;   // 8 waves/block, 1 pair/wave
        soap_gram_kernel<<<blocks, 256, 0, stream>>>(coeffs, (float*)d_out,
                                                     n_pairs);
    }
}